// EntropyRegularizationLoss_49383533969545
// MI455X (gfx1250) — compile-verified
//
#include <hip/hip_runtime.h>
#include <hip/hip_bf16.h>

// Problem constants (from reference: C=8192 classes, E=512 embed dim)
#define CROWS 8192
#define EDIM  512
#define NCHUNK 4                      // split j-range across blocks for occupancy
#define JT_TOTAL (CROWS / 16)         // 512 column tiles of 16
#define NJ 4                          // j-tiles per wave pass (register blocking)
#define JG_TOTAL (JT_TOTAL / NJ)      // 128 j-groups of 64 columns
#define JG_PER   (JG_TOTAL / NCHUNK)  // 32 j-groups per chunk
#define WAVES_PER_BLOCK 8
#define KC 128                        // K-chunk staged in LDS
#define BSTRIDE 132                   // 128 + 4 pad: 132 % 64 == 4 -> conflict-free

typedef float v2f __attribute__((ext_vector_type(2)));
typedef float v8f __attribute__((ext_vector_type(8)));

// ---------------------------------------------------------------------------
// Kernel 1: row L2-normalize.  One block (256 threads) per row of 512 floats.
// ---------------------------------------------------------------------------
__global__ void k_normalize(const float* __restrict__ x, float* __restrict__ pn) {
    const int row = blockIdx.x;
    const int tid = threadIdx.x;
    const float* xr = x + (size_t)row * EDIM;
    float v0 = xr[tid];
    float v1 = xr[tid + 256];
    float ss = v0 * v0 + v1 * v1;
    #pragma unroll
    for (int off = 16; off; off >>= 1) ss += __shfl_xor(ss, off, 32);
    __shared__ float red[8];
    if ((tid & 31) == 0) red[tid >> 5] = ss;
    __syncthreads();
    float tot = 0.f;
    #pragma unroll
    for (int w = 0; w < 8; ++w) tot += red[w];
    const float inv = 1.0f / fmaxf(sqrtf(tot), 1e-12f);
    float* pr = pn + (size_t)row * EDIM;
    pr[tid]       = v0 * inv;
    pr[tid + 256] = v1 * inv;
}

// ---------------------------------------------------------------------------
// Kernel 2: q[e] = sum_i pn[i][e]   (512 outputs, coalesced column walk)
// ---------------------------------------------------------------------------
__global__ void k_qsum(const float* __restrict__ pn, float* __restrict__ q) {
    const int e = blockIdx.x * blockDim.x + threadIdx.x;   // 0..511
    float s = 0.f;
    for (int i = 0; i < CROWS; ++i) s += pn[(size_t)i * EDIM + e];
    q[e] = s;
}

// ---------------------------------------------------------------------------
// Kernel 3: fused Gram + row-entropy, fixed softmax shift m=1 (|G_ij| <= 1
//   since rows are unit vectors), so state per row is just (s, t):
//      s = sum exp(x-1),  t = sum x*exp(x-1),  entropy = 1 + log s - t/s.
//   wave -> one 16-row i-tile x 4 j-tiles (NJ register blocking);
//   B block (64 rows x KC) staged in LDS per k-chunk, shared by 8 waves;
//   A fragments batched 4-deep from global (L2-resident) per load group.
// ---------------------------------------------------------------------------
__global__ void __launch_bounds__(256)
k_gram_entropy(const float* __restrict__ pn, float* __restrict__ states) {
    __shared__ __align__(16) float lb[16 * NJ * BSTRIDE];   // 64 x 132 f32 = 33 KB

    const int wave = threadIdx.x >> 5;
    const int lane = threadIdx.x & 31;
    const int lm = lane & 15;     // N / M index within fragment
    const int lh = lane >> 4;     // K half selector
    const int itile = blockIdx.x * WAVES_PER_BLOCK + wave;   // 0..511
    const int ibase = itile * 16;
    const int chunk = blockIdx.y;                            // 0..NCHUNK-1

    // A fragment source: lane reads pn[ibase + lm][k + 2*lh .. +1]  (float2)
    const float* arow = pn + (size_t)(ibase + lm) * EDIM + 2 * lh;

    float sacc[8], tacc[8];
    #pragma unroll
    for (int r = 0; r < 8; ++r) { sacc[r] = 0.f; tacc[r] = 0.f; }

    const int jg_begin = chunk * JG_PER;
    const int jg_end   = jg_begin + JG_PER;

    for (int jg = jg_begin; jg < jg_end; ++jg) {
        const int jbase = jg * (16 * NJ);     // first of 64 columns

        v8f acc[NJ];
        #pragma unroll
        for (int n = 0; n < NJ; ++n) acc[n] = {};

        for (int kc = 0; kc < EDIM; kc += KC) {
            // ---- stage B block (64 rows x KC f32) into padded LDS ----
            __syncthreads();
            for (int idx = threadIdx.x; idx < 16 * NJ * (KC / 4); idx += 256) {
                const int br = idx >> 5;              // 0..63   (KC/4 == 32)
                const int bc = (idx & 31) << 2;       // 0..124 step 4
                const float4 v = *(const float4*)(pn + (size_t)(jbase + br) * EDIM + kc + bc);
                *(float4*)(&lb[br * BSTRIDE + bc]) = v;
            }
            __syncthreads();

            const float* ap = arow + kc;
            #pragma unroll
            for (int kk = 0; kk < KC; kk += 16) {
                // 4 A fragments in flight (16 consecutive K values)
                const v2f a0 = *(const v2f*)(ap + kk);
                const v2f a1 = *(const v2f*)(ap + kk + 4);
                const v2f a2 = *(const v2f*)(ap + kk + 8);
                const v2f a3 = *(const v2f*)(ap + kk + 12);
                #pragma unroll
                for (int n = 0; n < NJ; ++n) {
                    const float* bp = &lb[(16 * n + lm) * BSTRIDE + 2 * lh + kk];
                    const v2f b0 = *(const v2f*)(bp);
                    const v2f b1 = *(const v2f*)(bp + 4);
                    const v2f b2 = *(const v2f*)(bp + 8);
                    const v2f b3 = *(const v2f*)(bp + 12);
                    acc[n] = __builtin_amdgcn_wmma_f32_16x16x4_f32(
                        false, a0, false, b0, (short)0, acc[n], false, false);
                    acc[n] = __builtin_amdgcn_wmma_f32_16x16x4_f32(
                        false, a1, false, b1, (short)0, acc[n], false, false);
                    acc[n] = __builtin_amdgcn_wmma_f32_16x16x4_f32(
                        false, a2, false, b2, (short)0, acc[n], false, false);
                    acc[n] = __builtin_amdgcn_wmma_f32_16x16x4_f32(
                        false, a3, false, b3, (short)0, acc[n], false, false);
                }
            }
        }

        // ---- entropy state update: acc[n][r] = G[ibase+r+8*lh][jbase+16n+lm]
        #pragma unroll
        for (int n = 0; n < NJ; ++n) {
            #pragma unroll
            for (int r = 0; r < 8; ++r) {
                const float x = acc[n][r];
                const float e = __expf(x - 1.0f);
                sacc[r] += e;
                tacc[r] += x * e;
            }
        }
    }

    // ---- sum across the 16 lanes of each half (plain additive merge) ----
    #pragma unroll
    for (int r = 0; r < 8; ++r) {
        float ss = sacc[r], tt = tacc[r];
        #pragma unroll
        for (int off = 1; off < 16; off <<= 1) {   // xor<16 stays inside half
            ss += __shfl_xor(ss, off, 32);
            tt += __shfl_xor(tt, off, 32);
        }
        if (lm == 0) {
            const int row = ibase + r + 8 * lh;
            float* st = states + ((size_t)row * NCHUNK + chunk) * 2;
            st[0] = ss; st[1] = tt;
        }
    }
}

// ---------------------------------------------------------------------------
// Kernel 4: colsum[j] = dot(q, pn[j])   (one wave per row)
// ---------------------------------------------------------------------------
__global__ void k_colsum(const float* __restrict__ pn, const float* __restrict__ q,
                         float* __restrict__ colsum) {
    const int wave = threadIdx.x >> 5;
    const int lane = threadIdx.x & 31;
    const int row  = blockIdx.x * 8 + wave;
    const float* pr = pn + (size_t)row * EDIM;
    float acc = 0.f;
    for (int k = lane; k < EDIM; k += 32) acc += q[k] * pr[k];
    #pragma unroll
    for (int off = 16; off; off >>= 1) acc += __shfl_xor(acc, off, 32);
    if (lane == 0) colsum[row] = acc;
}

// ---------------------------------------------------------------------------
// Kernel 5: sum per-row (s,t) chunk partials -> row entropies; online
//           entropy(colsum); out[0] = rowent/C + colent.  Deterministic.
// ---------------------------------------------------------------------------
__global__ void k_finalize(const float* __restrict__ states,
                           const float* __restrict__ colsum,
                           float* __restrict__ out) {
    const int tid  = threadIdx.x;
    const int lane = tid & 31;
    const int wv   = tid >> 5;

    // ---- part 1: per-row entropies (fixed shift m = 1) ----
    float rsum = 0.f;
    for (int row = tid; row < CROWS; row += 256) {
        const float* st = states + (size_t)row * NCHUNK * 2;
        float ss = 0.f, tt = 0.f;
        #pragma unroll
        for (int c = 0; c < NCHUNK; ++c) { ss += st[c * 2]; tt += st[c * 2 + 1]; }
        rsum += (1.0f + logf(ss)) - tt / ss;     // entropy = lse - E_p[x]
    }
    #pragma unroll
    for (int off = 16; off; off >>= 1) rsum += __shfl_xor(rsum, off, 32);
    __shared__ float red[8];
    if (lane == 0) red[wv] = rsum;

    // ---- part 2: online entropy of the 8192-long colsum vector ----
    float mm = -3.4e38f, ss = 0.f, tt = 0.f;
    for (int j = tid; j < CROWS; j += 256) {
        const float x  = colsum[j];
        const float nm = fmaxf(mm, x);
        const float e0 = __expf(mm - nm);
        const float e1 = __expf(x - nm);
        ss = ss * e0 + e1;
        tt = tt * e0 + x * e1;
        mm = nm;
    }
    #pragma unroll
    for (int off = 16; off; off >>= 1) {
        const float om = __shfl_xor(mm, off, 32);
        const float os = __shfl_xor(ss, off, 32);
        const float ot = __shfl_xor(tt, off, 32);
        const float nm = fmaxf(mm, om);
        const float e0 = __expf(mm - nm);
        const float e1 = __expf(om - nm);
        ss = ss * e0 + os * e1;
        tt = tt * e0 + ot * e1;
        mm = nm;
    }
    __shared__ float rm[8], rs[8], rt[8];
    if (lane == 0) { rm[wv] = mm; rs[wv] = ss; rt[wv] = tt; }
    __syncthreads();

    if (tid == 0) {
        float rowent = 0.f;
        #pragma unroll
        for (int w = 0; w < 8; ++w) rowent += red[w];
        float M = rm[0], S = rs[0], T = rt[0];
        for (int w = 1; w < 8; ++w) {
            const float nm = fmaxf(M, rm[w]);
            const float e0 = __expf(M - nm);
            const float e1 = __expf(rm[w] - nm);
            S = S * e0 + rs[w] * e1;
            T = T * e0 + rt[w] * e1;
            M = nm;
        }
        const float colent = (M + logf(S)) - T / S;
        out[0] = rowent / (float)CROWS + colent;
    }
}

// ---------------------------------------------------------------------------
extern "C" void kernel_launch(void* const* d_in, const int* in_sizes, int n_in,
                              void* d_out, int out_size, void* d_ws, size_t ws_size,
                              hipStream_t stream) {
    (void)in_sizes; (void)n_in; (void)out_size; (void)ws_size;
    const float* x = (const float*)d_in[0];     // prototypes [8192, 512] f32
    float* out = (float*)d_out;                 // scalar f32

    // workspace layout (floats): pn[C*E] | q[E] | colsum[C] | states[C*NCHUNK*2]
    float* pn     = (float*)d_ws;
    float* q      = pn + (size_t)CROWS * EDIM;
    float* colsum = q + EDIM;
    float* states = colsum + CROWS;

    k_normalize<<<CROWS, 256, 0, stream>>>(x, pn);
    k_qsum<<<EDIM / 256, 256, 0, stream>>>(pn, q);
    dim3 grid(JT_TOTAL / WAVES_PER_BLOCK, NCHUNK);   // (64, 4) -> 2048 waves
    k_gram_entropy<<<grid, 256, 0, stream>>>(pn, states);
    k_colsum<<<CROWS / 8, 256, 0, stream>>>(pn, q, colsum);
    k_finalize<<<1, 256, 0, stream>>>(states, colsum, out);
}